// Multi_GCNEncoder_55594056680018
// MI455X (gfx1250) — compile-verified
//
#include <hip/hip_runtime.h>
#include <math.h>

#define NCELLS 8192
#define NINPUT 2000
#define HDIM   128
#define NOUT   32
#define BN_EPS 1e-3f

typedef float v2f __attribute__((ext_vector_type(2)));
typedef float v8f __attribute__((ext_vector_type(8)));
typedef int   vi4 __attribute__((vector_size(16)));

#define BM 64
#define BN 64
#define BK 32
#define A_STRIDE 36   // padded LDS row stride (floats): v2f reads stay 8B-aligned
#define B_STRIDE 68   // padded LDS row stride (floats): float4 stores stay 16B-aligned

// CDNA5 async global->LDS copies (ASYNCcnt). Guarded so the source always compiles;
// fallback is global->VGPR->ds_store staging.
#if defined(__HIP_DEVICE_COMPILE__) && defined(__gfx1250__) &&                      \
    __has_builtin(__builtin_amdgcn_global_load_async_to_lds_b128) &&                \
    __has_builtin(__builtin_amdgcn_global_load_async_to_lds_b32) &&                 \
    __has_builtin(__builtin_amdgcn_s_wait_asynccnt)
#define USE_ASYNC_LDS 1
#else
#define USE_ASYNC_LDS 0
#endif

__device__ __forceinline__ void async_cp128(const float* g, float* l) {
#if USE_ASYNC_LDS
    __builtin_amdgcn_global_load_async_to_lds_b128(
        (__attribute__((address_space(1))) vi4*)(void*)g,
        (__attribute__((address_space(3))) vi4*)(void*)l, 0, 0);
#else
    *(float4*)l = *(const float4*)g;
#endif
}

__device__ __forceinline__ void async_cp32(const float* g, float* l) {
#if USE_ASYNC_LDS
    __builtin_amdgcn_global_load_async_to_lds_b32(
        (__attribute__((address_space(1))) int*)(void*)g,
        (__attribute__((address_space(3))) int*)(void*)l, 0, 0);
#else
    *l = *g;
#endif
}

__device__ __forceinline__ void async_fence() {
#if USE_ASYNC_LDS
    __builtin_amdgcn_s_wait_asynccnt(0);
#endif
}

// ---- tile staging -----------------------------------------------------------

__device__ __forceinline__ void stage_A(const float* __restrict__ A, int lda,
                                        int m0, int k0, int K, float* As_, int t)
{
    const int r  = t >> 3;          // 0..31
    const int c4 = (t & 7) << 2;    // 0..28
#pragma unroll
    for (int p = 0; p < 2; ++p) {
        const int row = p * 32 + r;
        const float* src = A + (long long)(m0 + row) * lda + k0 + c4;
        float* dst = &As_[row * A_STRIDE + c4];
        if (k0 + c4 + 3 < K) {
            async_cp128(src, dst);
        } else {
            float4 v;
            v.x = (k0 + c4 + 0 < K) ? src[0] : 0.f;
            v.y = (k0 + c4 + 1 < K) ? src[1] : 0.f;
            v.z = (k0 + c4 + 2 < K) ? src[2] : 0.f;
            v.w = (k0 + c4 + 3 < K) ? src[3] : 0.f;
            *(float4*)dst = v;
        }
    }
}

__device__ __forceinline__ void stage_Bn(const float* __restrict__ Bp, int ldb,
                                         int n0, int k0, int K, int N, float* Bs_, int t)
{
    const int r  = t >> 4;           // 0..15
    const int c4 = (t & 15) << 2;    // 0..60
#pragma unroll
    for (int p = 0; p < 2; ++p) {
        const int row = p * 16 + r;
        const int kk = k0 + row;
        const int nn = n0 + c4;
        const float* src = Bp + (long long)kk * ldb + nn;
        float* dst = &Bs_[row * B_STRIDE + c4];
        if (kk < K && nn + 3 < N) {
            async_cp128(src, dst);
        } else {
            float4 v = {0.f, 0.f, 0.f, 0.f};
            if (kk < K) {
                v.x = (nn + 0 < N) ? src[0] : 0.f;
                v.y = (nn + 1 < N) ? src[1] : 0.f;
                v.z = (nn + 2 < N) ? src[2] : 0.f;
                v.w = (nn + 3 < N) ? src[3] : 0.f;
            }
            *(float4*)dst = v;
        }
    }
}

// B^T staging: Bt is [N x K] row-major; Bs[k][n] = Bt[n0+n][k0+k].
// Async works here too: both the global and LDS addresses are per-lane.
__device__ __forceinline__ void stage_Bt(const float* __restrict__ Bp, int ldb,
                                         int n0, int k0, int K, int N, float* Bs_, int t)
{
    const int n  = t >> 2;           // 0..63
    const int kc = (t & 3) << 3;     // 0,8,16,24
    const float* src = Bp + (long long)(n0 + n) * ldb + k0 + kc;
#pragma unroll
    for (int q = 0; q < 2; ++q) {
        const int kq = kc + q * 4;
        if ((n0 + n) < N && (k0 + kq + 3) < K) {
#pragma unroll
            for (int i = 0; i < 4; ++i)
                async_cp32(src + q * 4 + i, &Bs_[(kq + i) * B_STRIDE + n]);
        } else {
#pragma unroll
            for (int i = 0; i < 4; ++i) {
                float v = ((n0 + n) < N && (k0 + kq + i) < K) ? src[q * 4 + i] : 0.f;
                Bs_[(kq + i) * B_STRIDE + n] = v;
            }
        }
    }
}

// Prefetch all fragments for a BK tile into registers, then run the WMMA burst.
// Keeps the ds_load latency out of the WMMA dependency chain (waits retire with
// dscnt>0 instead of per-pair s_wait_dscnt 0).
__device__ __forceinline__ void compute_tile(const float* As_, const float* Bs_,
                                             int wm, int wn, int trow, int koff,
                                             v8f& acc0, v8f& acc1)
{
    v2f af0[8], af1[8], bf[8];
#pragma unroll
    for (int s = 0; s < 8; ++s) {
        const int kk = s * 4;
        af0[s] = *(const v2f*)&As_[(wm * 32 +      trow) * A_STRIDE + kk + koff];
        af1[s] = *(const v2f*)&As_[(wm * 32 + 16 + trow) * A_STRIDE + kk + koff];
        bf[s].x = Bs_[(kk + koff    ) * B_STRIDE + wn * 16 + trow];
        bf[s].y = Bs_[(kk + koff + 1) * B_STRIDE + wn * 16 + trow];
    }
#pragma unroll
    for (int s = 0; s < 8; ++s) {
        acc0 = __builtin_amdgcn_wmma_f32_16x16x4_f32(false, af0[s], false, bf[s], (short)0, acc0, false, false);
        acc1 = __builtin_amdgcn_wmma_f32_16x16x4_f32(false, af1[s], false, bf[s], (short)0, acc1, false, false);
    }
}

// Generic C = alpha * (A @ B) + bias, fp32, V_WMMA_F32_16X16X4_F32,
// double-buffered async global->LDS staging. opB=1: B is B^T [N x K] row-major.
__global__ __launch_bounds__(256)
void gemm_f32_wmma(const float* __restrict__ A, int lda,
                   const float* __restrict__ B, int ldb, int opB,
                   const float* __restrict__ bias,
                   float* __restrict__ C, int ldc,
                   int M, int N, int K, float alpha,
                   const int* __restrict__ head_ptr,
                   long long b_head_stride, int bias_head_stride)
{
    __shared__ __align__(16) float As0[BM * A_STRIDE];
    __shared__ __align__(16) float As1[BM * A_STRIDE];
    __shared__ __align__(16) float Bs0[BK * B_STRIDE];
    __shared__ __align__(16) float Bs1[BK * B_STRIDE];

    const int t    = threadIdx.x;
    const int lane = t & 31;
    const int wave = t >> 5;      // 0..7
    const int wm   = wave >> 2;   // 0..1  (32 rows each)
    const int wn   = wave & 3;    // 0..3  (16 cols each)

    const int m0 = blockIdx.y * BM;
    const int n0 = blockIdx.x * BN;

    const int head = head_ptr ? head_ptr[0] : 0;
    const float* Bp    = B + (long long)head * b_head_stride;
    const float* biasp = bias ? (bias + head * bias_head_stride) : nullptr;

    v8f acc0 = {};  // rows [wm*32,    wm*32+16)
    v8f acc1 = {};  // rows [wm*32+16, wm*32+32)

    const int trow = lane & 15;          // M (A) / N (B) within 16x16 subtile
    const int koff = (lane >> 4) << 1;   // 0 or 2 (f32 WMMA K striping)

    const int nk = (K + BK - 1) / BK;

    // prologue: stage tile 0 into buffer 0
    stage_A(A, lda, m0, 0, K, As0, t);
    if (!opB) stage_Bn(Bp, ldb, n0, 0, K, N, Bs0, t);
    else      stage_Bt(Bp, ldb, n0, 0, K, N, Bs0, t);
    async_fence();
    __syncthreads();

    for (int i = 0; i < nk; ++i) {
        const float* Asb = (i & 1) ? As1 : As0;
        const float* Bsb = (i & 1) ? Bs1 : Bs0;
        float* Asn = (i & 1) ? As0 : As1;
        float* Bsn = (i & 1) ? Bs0 : Bs1;
        if (i + 1 < nk) {
            const int k0 = (i + 1) * BK;
            stage_A(A, lda, m0, k0, K, Asn, t);
            if (!opB) stage_Bn(Bp, ldb, n0, k0, K, N, Bsn, t);
            else      stage_Bt(Bp, ldb, n0, k0, K, N, Bsn, t);
        }
        // WMMA on the current buffers overlaps the in-flight async copies
        compute_tile(Asb, Bsb, wm, wn, trow, koff, acc0, acc1);
        async_fence();
        __syncthreads();
    }

    // epilogue: C/D layout: VGPR r -> m = r (+8 for lanes 16..31), n = lane&15
    const int cn = n0 + wn * 16 + (lane & 15);
    if (cn < N) {
        const float bv = biasp ? biasp[cn] : 0.f;
        const int rbase = m0 + wm * 32 + ((lane >> 4) << 3);
#pragma unroll
        for (int r = 0; r < 8; ++r) {
            C[(long long)(rbase + r     ) * ldc + cn] = alpha * acc0[r] + bv;
            C[(long long)(rbase + 16 + r) * ldc + cn] = alpha * acc1[r] + bv;
        }
    }
}

// per-column batch mean / biased variance (BatchNorm1d training stats)
__global__ __launch_bounds__(256)
void col_stats(const float* __restrict__ h, int Nrows, int C,
               float* __restrict__ mu, float* __restrict__ var)
{
    __shared__ float s1[256];
    __shared__ float s2[256];
    const int c = blockIdx.x;
    float sum = 0.f, sq = 0.f;
    for (int r = threadIdx.x; r < Nrows; r += 256) {
        float v = h[(long long)r * C + c];
        sum += v; sq += v * v;
    }
    s1[threadIdx.x] = sum; s2[threadIdx.x] = sq;
    __syncthreads();
    for (int s = 128; s > 0; s >>= 1) {
        if (threadIdx.x < s) {
            s1[threadIdx.x] += s1[threadIdx.x + s];
            s2[threadIdx.x] += s2[threadIdx.x + s];
        }
        __syncthreads();
    }
    if (threadIdx.x == 0) {
        float m = s1[0] / (float)Nrows;
        mu[c]  = m;
        var[c] = s2[0] / (float)Nrows - m * m;
    }
}

__global__ __launch_bounds__(256)
void bn_relu(const float* __restrict__ h, float* __restrict__ out,
             const float* __restrict__ mu, const float* __restrict__ var,
             const float* __restrict__ g, const float* __restrict__ be,
             const int* __restrict__ head_ptr, int head_stride, int total, int C)
{
    const int i = blockIdx.x * blockDim.x + threadIdx.x;
    if (i >= total) return;
    const int c = i & (C - 1);   // C is a power of two (128)
    const int head = head_ptr ? head_ptr[0] : 0;
    const int gi = head * head_stride + c;
    float v = (h[i] - mu[c]) * rsqrtf(var[c] + BN_EPS) * g[gi] + be[gi];
    out[i] = v > 0.f ? v : 0.f;
}

// row softmax, in place; whole 32KB row cached in LDS -> one HBM read + one write
__global__ __launch_bounds__(256)
void softmax_rows_8192(float* __restrict__ attn)
{
    __shared__ __align__(16) float row[8192];
    __shared__ float red[256];
    const int t = threadIdx.x;
    float* p = attn + (long long)blockIdx.x * 8192;

    float mx = -3.402823466e38f;
    for (int i = t * 4; i < 8192; i += 1024) {
        float4 v = *(const float4*)(p + i);
        *(float4*)&row[i] = v;
        mx = fmaxf(mx, fmaxf(fmaxf(v.x, v.y), fmaxf(v.z, v.w)));
    }
    red[t] = mx; __syncthreads();
    for (int s = 128; s > 0; s >>= 1) {
        if (t < s) red[t] = fmaxf(red[t], red[t + s]);
        __syncthreads();
    }
    mx = red[0];
    __syncthreads();

    float sum = 0.f;
    for (int i = t * 4; i < 8192; i += 1024) {
        float4 v = *(float4*)&row[i];
        v.x = expf(v.x - mx); v.y = expf(v.y - mx);
        v.z = expf(v.z - mx); v.w = expf(v.w - mx);
        sum += v.x + v.y + v.z + v.w;
        *(float4*)&row[i] = v;
    }
    red[t] = sum; __syncthreads();
    for (int s = 128; s > 0; s >>= 1) {
        if (t < s) red[t] += red[t + s];
        __syncthreads();
    }
    const float rinv = 1.f / red[0];
    for (int i = t * 4; i < 8192; i += 1024) {
        float4 v = *(float4*)&row[i];
        v.x *= rinv; v.y *= rinv; v.z *= rinv; v.w *= rinv;
        *(float4*)(p + i) = v;
    }
}

__global__ __launch_bounds__(256)
void finalize_latent(const float* __restrict__ qm, const float* __restrict__ tvar,
                     const float* __restrict__ eps, float* __restrict__ qv,
                     float* __restrict__ lat, int n)
{
    const int i = blockIdx.x * blockDim.x + threadIdx.x;
    if (i >= n) return;
    float v = expf(tvar[i]);
    qv[i]  = v;
    lat[i] = qm[i] + sqrtf(v) * eps[i];
}

extern "C" void kernel_launch(void* const* d_in, const int* in_sizes, int n_in,
                              void* d_out, int out_size, void* d_ws, size_t ws_size,
                              hipStream_t stream)
{
    (void)in_sizes; (void)n_in; (void)out_size; (void)ws_size;

    const float* x      = (const float*)d_in[0];
    const float* spat   = (const float*)d_in[1];
    const float* eps    = (const float*)d_in[2];
    const float* enc_W  = (const float*)d_in[3];
    const float* enc_b  = (const float*)d_in[4];
    const float* enc_g  = (const float*)d_in[5];
    const float* enc_be = (const float*)d_in[6];
    const float* sh_W1  = (const float*)d_in[7];
    const float* sh_b1  = (const float*)d_in[8];
    const float* sh_g1  = (const float*)d_in[9];
    const float* sh_be1 = (const float*)d_in[10];
    const float* sh_W2  = (const float*)d_in[11];
    const float* sh_b2  = (const float*)d_in[12];
    const float* sh_g2  = (const float*)d_in[13];
    const float* sh_be2 = (const float*)d_in[14];
    const float* qkv_W  = (const float*)d_in[15];
    const float* qkv_b  = (const float*)d_in[16];
    const float* o_W    = (const float*)d_in[17];
    const float* o_b    = (const float*)d_in[18];
    const float* mean_W = (const float*)d_in[19];
    const float* mean_b = (const float*)d_in[20];
    const float* var_W  = (const float*)d_in[21];
    const float* var_b  = (const float*)d_in[22];
    const int*   head   = (const int*)d_in[23];

    const long long NH = (long long)NCELLS * HDIM;
    float* ws   = (float*)d_ws;
    float* h1   = ws;                 // [N,H]
    float* h2   = ws + NH;            // [N,H]
    float* qkv  = ws + 2 * NH;        // [N,3H]
    float* kagg = ws + 5 * NH;        // [N,H]
    float* av   = h1;                 // reuse after qkv is built
    float* o    = h2;                 // reuse
    float* tvar = ws + 6 * NH;        // [N,NOUT]
    float* mu   = tvar + (long long)NCELLS * NOUT;
    float* var  = mu + 512;

    float* outp = (float*)d_out;
    float* qm   = outp;
    float* qv   = outp + (long long)NCELLS * NOUT;
    float* lat  = outp + 2LL * NCELLS * NOUT;
    float* attn = outp + 3LL * NCELLS * NOUT;

    const dim3 blk(256);
    const float inv_sqrt_h = 0.08838834764831845f;  // 1/sqrt(128)
    const int nih = (NCELLS * HDIM + 255) / 256;

    #define GEMM(A, lda, B, ldb, opB, bias, C, ldc, M, Nn, K, alpha, hp, bhs, bih)        \
        gemm_f32_wmma<<<dim3(((Nn) + 63) / 64, ((M) + 63) / 64), blk, 0, stream>>>(       \
            A, lda, B, ldb, opB, bias, C, ldc, M, Nn, K, alpha, hp, bhs, bih)

    // 1) head-specific encoder: h1 = relu(BN(x @ enc_W[head] + enc_b[head]))
    GEMM(x, NINPUT, enc_W, HDIM, 0, enc_b, h1, HDIM, NCELLS, HDIM, NINPUT, 1.f,
         head, (long long)NINPUT * HDIM, HDIM);
    col_stats<<<dim3(HDIM), blk, 0, stream>>>(h1, NCELLS, HDIM, mu, var);
    bn_relu<<<dim3(nih), blk, 0, stream>>>(h1, h1, mu, var, enc_g, enc_be, head, HDIM,
                                           NCELLS * HDIM, HDIM);
    // 2) shared layer 1
    GEMM(h1, HDIM, sh_W1, HDIM, 0, sh_b1, h2, HDIM, NCELLS, HDIM, HDIM, 1.f,
         (const int*)nullptr, 0LL, 0);
    col_stats<<<dim3(HDIM), blk, 0, stream>>>(h2, NCELLS, HDIM, mu, var);
    bn_relu<<<dim3(nih), blk, 0, stream>>>(h2, h2, mu, var, sh_g1, sh_be1,
                                           (const int*)nullptr, 0, NCELLS * HDIM, HDIM);
    // 3) shared layer 2
    GEMM(h2, HDIM, sh_W2, HDIM, 0, sh_b2, h1, HDIM, NCELLS, HDIM, HDIM, 1.f,
         (const int*)nullptr, 0LL, 0);
    col_stats<<<dim3(HDIM), blk, 0, stream>>>(h1, NCELLS, HDIM, mu, var);
    bn_relu<<<dim3(nih), blk, 0, stream>>>(h1, h1, mu, var, sh_g2, sh_be2,
                                           (const int*)nullptr, 0, NCELLS * HDIM, HDIM);
    // 4) qkv = h1 @ qkv_W + qkv_b   (q|k|v packed, ld = 3H)
    GEMM(h1, HDIM, qkv_W, 3 * HDIM, 0, qkv_b, qkv, 3 * HDIM, NCELLS, 3 * HDIM, HDIM, 1.f,
         (const int*)nullptr, 0LL, 0);
    // 5) k_agg = spatial_metric @ k
    GEMM(spat, NCELLS, qkv + HDIM, 3 * HDIM, 0, (const float*)nullptr, kagg, HDIM,
         NCELLS, HDIM, NCELLS, 1.f, (const int*)nullptr, 0LL, 0);
    // 6) logits = (q @ k_agg^T) / sqrt(H), written directly into the attn output slot
    GEMM(qkv, 3 * HDIM, kagg, HDIM, 1, (const float*)nullptr, attn, NCELLS,
         NCELLS, NCELLS, HDIM, inv_sqrt_h, (const int*)nullptr, 0LL, 0);
    // 7) softmax rows, in place
    softmax_rows_8192<<<dim3(NCELLS), blk, 0, stream>>>(attn);
    // 8) av = attn @ v
    GEMM(attn, NCELLS, qkv + 2 * HDIM, 3 * HDIM, 0, (const float*)nullptr, av, HDIM,
         NCELLS, HDIM, NCELLS, 1.f, (const int*)nullptr, 0LL, 0);
    // 9) o = av @ o_W + o_b
    GEMM(av, HDIM, o_W, HDIM, 0, o_b, o, HDIM, NCELLS, HDIM, HDIM, 1.f,
         (const int*)nullptr, 0LL, 0);
    // 10) q_m = o @ mean_W + mean_b  (straight into d_out)
    GEMM(o, HDIM, mean_W, NOUT, 0, mean_b, qm, NOUT, NCELLS, NOUT, HDIM, 1.f,
         (const int*)nullptr, 0LL, 0);
    // 11) tvar = o @ var_W + var_b
    GEMM(o, HDIM, var_W, NOUT, 0, var_b, tvar, NOUT, NCELLS, NOUT, HDIM, 1.f,
         (const int*)nullptr, 0LL, 0);
    // 12) q_v = exp(tvar); latent = q_m + sqrt(q_v) * eps
    finalize_latent<<<dim3((NCELLS * NOUT + 255) / 256), blk, 0, stream>>>(
        qm, tvar, eps, qv, lat, NCELLS * NOUT);

    #undef GEMM
}